// Program_4578435138231
// MI455X (gfx1250) — compile-verified
//
#include <hip/hip_runtime.h>

// MI455X / gfx1250, wave32. Matrix ops via V_WMMA_F32_16X16X4_F32 (exact f32).
// Each 32-lane wave processes 32 rows (two 16-row WMMA tiles).
// Layer1:  h3^T[32x16] = l1w^T[32x4] @ h2^T[4x16] + l1b   (2 WMMA / tile)
// Layer2:  p^T[16x16]  = l2w^T[16x32] @ h3^T[32x16] + l2b (8 chained WMMA / tile)
// Layer1's D layout feeds layer2's B with one ds_bpermute + cndmask per VGPR.

typedef __attribute__((ext_vector_type(2))) float v2f;
typedef __attribute__((ext_vector_type(8))) float v8f;

#define SPEED 5.0f

__device__ __forceinline__ float bperm(int idx_bytes, float v) {
  // ds_bpermute_b32: dst[lane] = src[idx[lane]/4]  (convergent)
  return __builtin_bit_cast(
      float, __builtin_amdgcn_ds_bpermute(idx_bytes, __builtin_bit_cast(int, v)));
}

__device__ __forceinline__ v8f wmma4(v2f a, v2f b, v8f c) {
  // D[16x16] = A[16x4] * B[4x16] + C   (f32, exact)
  return __builtin_amdgcn_wmma_f32_16x16x4_f32(
      /*neg_a=*/false, a, /*neg_b=*/false, b,
      /*c_mod=*/(short)0, c, /*reuse_a=*/false, /*reuse_b=*/false);
}

// Single v_max_num_f32 (select form; single op when input known canonical).
__device__ __forceinline__ float relu(float x) { return (x > 0.0f) ? x : 0.0f; }

// Raw single-instruction ReLU for WMMA outputs (avoids the canonicalize
// v_max x,x the backend inserts for opaque intrinsic results).
__device__ __forceinline__ float relu_raw(float x) {
  float r;
  asm("v_max_num_f32 %0, 0, %1" : "=v"(r) : "v"(x));
  return r;
}

__device__ __forceinline__ float sigmoid(float x) {
  // v_exp path + v_rcp_f32 (~1ulp), plenty for f32 tolerance.
  return __builtin_amdgcn_rcpf(1.0f + __expf(-x));
}

__device__ __forceinline__ void store2(float* p, float a, float b) {
  v2f v; v.x = a; v.y = b;
  *reinterpret_cast<v2f*>(p) = v;  // 8B-aligned by construction -> b64 store
}

__global__ __launch_bounds__(256) void sim_kernel(
    const float* __restrict__ x,
    const float* __restrict__ c1w, const float* __restrict__ c1b,
    const float* __restrict__ c2w, const float* __restrict__ c2b,
    const float* __restrict__ l1w, const float* __restrict__ l1b,
    const float* __restrict__ l2w, const float* __restrict__ l2b,
    float* __restrict__ out, int nrows) {
  const int tid = blockIdx.x * 256 + threadIdx.x;
  const int row = (tid < nrows) ? tid : (nrows - 1);
  const bool wr = (tid < nrows);

  const int lane  = threadIdx.x & 31;
  const int mloc  = lane & 15;   // position within half-wave
  const int khalf = lane >> 4;   // 0: K in {0,1}, 1: K in {2,3}
  const bool low  = (khalf == 0);
  const int idxLow  = mloc * 4;          // bperm byte index -> lane mloc
  const int idxHigh = (mloc + 16) * 4;   // bperm byte index -> lane mloc+16

  // uniform conv weights (scalar loads)
  const float w10 = c1w[0], w11 = c1w[1], cb1 = c1b[0];
  const float w20 = c2w[0], w21 = c2w[1], cb2 = c2b[0];

  // ---- constant WMMA operands, loaded once ----
  // Layer1 A = l1w^T (32x4) split into feature halves (M=0..15 / 16..31).
  // f32 A layout: element(M,K) -> VGPR K%2, lane M + 16*(K>=2).
  v2f A1lo, A1hi;
  A1lo.x = l1w[(2 * khalf + 0) * 32 + mloc];
  A1lo.y = l1w[(2 * khalf + 1) * 32 + mloc];
  A1hi.x = l1w[(2 * khalf + 0) * 32 + mloc + 16];
  A1hi.y = l1w[(2 * khalf + 1) * 32 + mloc + 16];

  // Layer1 bias as C matrix: element(m,n)=l1b[m]; C layout: VGPR m%8,
  // lanes 0-15 hold m=r, lanes 16-31 hold m=r+8.
  v8f C1lo, C1hi;
#pragma unroll
  for (int r = 0; r < 8; ++r) {
    C1lo[r] = l1b[r + 8 * khalf];
    C1hi[r] = l1b[16 + r + 8 * khalf];
  }

  // Layer2 A = l2w^T (5x32 padded to 16x32), 8 K-chunks of 4.
  v2f A2[8];
#pragma unroll
  for (int k = 0; k < 8; ++k) {
    A2[k].x = (mloc < 5) ? l2w[(4 * k + 2 * khalf + 0) * 5 + mloc] : 0.0f;
    A2[k].y = (mloc < 5) ? l2w[(4 * k + 2 * khalf + 1) * 5 + mloc] : 0.0f;
  }

  // Layer2 bias as C matrix (only m<5, lanes 0-15 region matters).
  v8f C2;
#pragma unroll
  for (int r = 0; r < 8; ++r) C2[r] = (low && r < 5) ? l2b[r] : 0.0f;

  // ---- per-row state ----
  float s[18];
#pragma unroll
  for (int j = 0; j < 18; ++j) s[j] = x[(size_t)row * 18 + j];

  {
    float u = s[1] - s[3], v = s[2] - s[4];
    s[10] = u * u + v * v;
  }

  float* orow = out + (size_t)row * 178;
  auto write_traj = [&](int t) {
    float* p_ = orow + 18 + t * 10;   // byte offset 72 + 40t from 8B-aligned row
    store2(p_ + 0, s[10], s[1]);
    store2(p_ + 2, s[2],  s[3]);
    store2(p_ + 4, s[4],  s[5]);
    store2(p_ + 6, s[6],  s[7]);
    store2(p_ + 8, s[8],  s[17]);
  };
  if (wr) write_traj(0);

#pragma unroll 1
  for (int t = 0; t < 15; ++t) {
    // ---- conv layers (per-lane scalar, weights uniform) ----
    float f0 = s[1], f1 = s[2], f2 = s[3], f3 = s[4], f4 = s[9], f5 = s[17];
    float h10 = relu(fmaf(w11, f1, fmaf(w10, f0, cb1)));
    float h11 = relu(fmaf(w11, f2, fmaf(w10, f1, cb1)));
    float h12 = relu(fmaf(w11, f3, fmaf(w10, f2, cb1)));
    float h13 = relu(fmaf(w11, f4, fmaf(w10, f3, cb1)));
    float h14 = relu(fmaf(w11, f5, fmaf(w10, f4, cb1)));
    float h20 = relu(fmaf(w21, h11, fmaf(w20, h10, cb2)));
    float h21 = relu(fmaf(w21, h12, fmaf(w20, h11, cb2)));
    float h22 = relu(fmaf(w21, h13, fmaf(w20, h12, cb2)));
    float h23 = relu(fmaf(w21, h14, fmaf(w20, h13, cb2)));

    // ---- B = h2^T (4x16) per tile; B layout: VGPR K%2, lane N+16*(K>=2) ----
    v2f B0, B1;
    {
      float ta = bperm(idxLow, h22), tb = bperm(idxLow, h23);
      B0.x = low ? h20 : ta;
      B0.y = low ? h21 : tb;
    }
    {
      float ta = bperm(idxHigh, h20), tb = bperm(idxHigh, h21);
      B1.x = low ? ta : h22;
      B1.y = low ? tb : h23;
    }

    // ---- layer1: h3^T = l1w^T @ h2^T + l1b (bias via C) ----
    v8f D0lo = wmma4(A1lo, B0, C1lo);
    v8f D0hi = wmma4(A1hi, B0, C1hi);
    v8f D1lo = wmma4(A1lo, B1, C1lo);
    v8f D1hi = wmma4(A1hi, B1, C1hi);
#pragma unroll
    for (int r = 0; r < 8; ++r) {
      D0lo[r] = relu_raw(D0lo[r]);
      D0hi[r] = relu_raw(D0hi[r]);
      D1lo[r] = relu_raw(D1lo[r]);
      D1hi[r] = relu_raw(D1hi[r]);
    }

    // ---- layer2: p^T = l2w^T @ h3^T + l2b, K=32 as 8 chained K=4 WMMAs.
    // D layout of layer1 feeds layer2's B with one bperm+select per VGPR.
    auto layer2 = [&](const v8f& Dlo, const v8f& Dhi) -> v8f {
      v8f acc = C2;
#pragma unroll
      for (int k = 0; k < 8; ++k) {
        const v8f& D = (k < 4) ? Dlo : Dhi;
        const int mm = (4 * k) & 15;  // 0,4,8,12
        v2f Bk;
        if (mm < 8) {
          // source rows m in lanes 0-15 of D
          float ta = bperm(idxLow, D[mm + 2]);
          float tb = bperm(idxLow, D[mm + 3]);
          Bk.x = low ? D[mm + 0] : ta;
          Bk.y = low ? D[mm + 1] : tb;
        } else {
          // source rows m in lanes 16-31 of D
          float ta = bperm(idxHigh, D[mm - 8 + 0]);
          float tb = bperm(idxHigh, D[mm - 8 + 1]);
          Bk.x = low ? ta : D[mm - 8 + 2];
          Bk.y = low ? tb : D[mm - 8 + 3];
        }
        acc = wmma4(A2[k], Bk, acc);
      }
      return acc;
    };
    v8f P0 = layer2(D0lo, D0hi);
    v8f P1 = layer2(D1lo, D1hi);

    // ---- gather this lane's 5 outputs: P element(j,n) -> VGPR j, lane n ----
    float p[5];
#pragma unroll
    for (int j = 0; j < 5; ++j) {
      float ta = bperm(idxLow, P1[j]);
      p[j] = low ? P0[j] : ta;
    }
#pragma unroll
    for (int j = 0; j < 5; ++j) p[j] = sigmoid(p[j]);

    // ---- branchy state update (per-lane scalar) ----
    s[5] = p[0]; s[6] = p[1]; s[7] = p[2]; s[8] = p[3]; s[17] = p[4];
    float a = p[1] - p[0], b = p[2] - p[0], c = p[3] - p[0];
    float d = p[2] - p[1], e = p[3] - p[1], f = p[3] - p[2];
    s[11] = a; s[12] = b; s[13] = c; s[14] = d; s[15] = e; s[16] = f;
    float dx_c = (c <= 0.0f) ? 0.0f : SPEED;
    float st_c = (c <= 0.0f) ? 0.0f : 3.0f;
    float dx_f = (f <= 0.0f) ? 0.0f : SPEED;
    float st_f = (f <= 0.0f) ? 2.0f : 3.0f;
    float dx_e = (e <= 0.0f) ? -SPEED : SPEED;
    float st_e = (e <= 0.0f) ? 1.0f : 3.0f;
    float dx_b = (b <= 0.0f) ? dx_c : dx_f;
    float st_b = (b <= 0.0f) ? st_c : st_f;
    float dx_d = (d <= 0.0f) ? dx_e : dx_f;
    float st_d = (d <= 0.0f) ? st_e : st_f;
    float dx = (a <= 0.0f) ? dx_b : dx_d;
    float st = (a <= 0.0f) ? st_b : st_d;
    s[1] += dx;
    s[9] = st;
    s[2] += SPEED; s[3] += SPEED; s[0] += 1.0f;
    float u = s[1] - s[3], v = s[2] - s[4];
    s[10] = u * u + v * v;

    if (wr) write_traj(t + 1);
  }

  if (wr) {
#pragma unroll
    for (int j = 0; j < 18; j += 2) store2(orow + j, s[j], s[j + 1]);
  }
}

extern "C" void kernel_launch(void* const* d_in, const int* in_sizes, int n_in,
                              void* d_out, int out_size, void* d_ws, size_t ws_size,
                              hipStream_t stream) {
  (void)n_in; (void)out_size; (void)d_ws; (void)ws_size;
  const float* x   = (const float*)d_in[0];
  const float* c1w = (const float*)d_in[1];
  const float* c1b = (const float*)d_in[2];
  const float* c2w = (const float*)d_in[3];
  const float* c2b = (const float*)d_in[4];
  const float* l1w = (const float*)d_in[5];
  const float* l1b = (const float*)d_in[6];
  const float* l2w = (const float*)d_in[7];
  const float* l2b = (const float*)d_in[8];
  float* out = (float*)d_out;

  const int nrows = in_sizes[0] / 18;
  const int threads = 256;
  const int blocks = (nrows + threads - 1) / threads;
  sim_kernel<<<blocks, threads, 0, stream>>>(x, c1w, c1b, c2w, c2b,
                                             l1w, l1b, l2w, l2b, out, nrows);
}